// Parser_67173288509811
// MI455X (gfx1250) — compile-verified
//
#include <hip/hip_runtime.h>
#include <hip/hip_bf16.h>
#include <math.h>

// ---------------- problem constants ----------------
constexpr int B_   = 32;
constexpr int W_   = 256;
constexpr int SUB_ = 512;
constexpr int D_   = 768;
constexpr int H_   = 10;
constexpr int G4_  = 40;   // 4*H
constexpr int NT_  = 80;   // both directions stacked: 5 WMMA n-tiles of 16
constexpr int HID_ = 30;
constexpr int HP_  = 31;   // HID + 1 (augmented)
constexpr int C_   = 50;
constexpr long LSTRIDE = (long)B_ * (SUB_ + 1) * D_;   // layer stride in hiddens

// workspace layout (floats)
constexpr long WORD_N = (long)B_ * W_ * D_;   // pooled word vectors
constexpr long CNT_N  = (long)B_ * W_;
constexpr long PRE_N  = (long)W_ * B_ * G4_;  // [t][b][g]
constexpr long U_N    = (long)B_ * W_ * 20;
constexpr long P32_N  = (long)B_ * W_ * 32;   // padded 31->32
constexpr long PICK_N = (long)B_ * W_ * C_;

constexpr long O_WORD = 0;
constexpr long O_CNT  = O_WORD + WORD_N;
constexpr long O_INV  = O_CNT  + CNT_N;
constexpr long O_PREF = O_INV  + CNT_N;
constexpr long O_PREB = O_PREF + PRE_N;
constexpr long O_U    = O_PREB + PRE_N;
constexpr long O_H1U  = O_U    + U_N;
constexpr long O_H2U  = O_H1U  + P32_N;
constexpr long O_H1R  = O_H2U  + P32_N;
constexpr long O_H2R  = O_H1R  + P32_N;
constexpr long O_G1U  = O_H2R  + P32_N;
constexpr long O_PICK = O_G1U  + P32_N;

typedef __attribute__((ext_vector_type(2))) float v2f;
typedef __attribute__((ext_vector_type(8))) float v8f;

__device__ __forceinline__ float sigmoidf_(float x) { return 1.0f / (1.0f + expf(-x)); }

// ---------------- K0: zero ----------------
__global__ void k_zero(float* __restrict__ p, long n) {
    long i = (long)blockIdx.x * blockDim.x + threadIdx.x;
    if (i < n) p[i] = 0.0f;
}

// ---------------- K1: mean over layers + segment-sum (general bert2toks) ----------------
// grid: B*SUB blocks, 256 threads; each block pools one subword row into its word slot.
__global__ __launch_bounds__(256) void k_segsum(const float* __restrict__ hiddens,
                                                const int* __restrict__ bert2toks,
                                                float* __restrict__ word,
                                                float* __restrict__ cnt) {
    int bs = blockIdx.x;
    int b  = bs >> 9;          // / SUB_
    int s  = bs & (SUB_ - 1);
    int w  = bert2toks[b * SUB_ + s];
    const float* base = hiddens + ((long)b * (SUB_ + 1) + (s + 1)) * D_; // drop token 0
    float* dst = word + ((long)b * W_ + w) * D_;
    for (int d = threadIdx.x; d < D_; d += 256) {
        float v = (base[d] + base[LSTRIDE + d] + base[2 * LSTRIDE + d]) * (1.0f / 3.0f);
        atomicAdd(&dst[d], v);
    }
    if (threadIdx.x == 0) atomicAdd(&cnt[b * W_ + w], 1.0f);
}

// ---------------- K1b: reciprocal counts ----------------
__global__ void k_inv(const float* __restrict__ cnt, float* __restrict__ inv) {
    int i = blockIdx.x * blockDim.x + threadIdx.x;
    if (i < B_ * W_) inv[i] = 1.0f / fmaxf(cnt[i], 1.0f);
}

// ---------------- K2: fused scale + input GEMM via V_WMMA_F32_16X16X4_F32 ----------------
// pre_all[m, n] = word[m,:]*inv[m] @ Wc[n,:]^T + bias[n], m = b*256+t, n in [0,80)
// one wave per 16-row M tile; 5 accumulators cover all 80 output columns.
__global__ __launch_bounds__(32) void k_pre_wmma(const float* __restrict__ word,
                                                 const float* __restrict__ inv,
                                                 const float* __restrict__ w_ih_f,
                                                 const float* __restrict__ w_ih_b,
                                                 const float* __restrict__ b_f,
                                                 const float* __restrict__ b_b,
                                                 float* __restrict__ pre_f,
                                                 float* __restrict__ pre_b) {
    const int m0   = blockIdx.x * 16;
    const int lane = threadIdx.x;
    const int half = lane >> 4;   // 0: K=0/1 , 1: K=2/3 (ISA 16x4 f32 A layout)
    const int l16  = lane & 15;
    const int arow = m0 + l16;
    const float rscale = inv[arow];
    const float* arp = word + (long)arow * D_;

    v8f acc[5];
    #pragma unroll
    for (int nt = 0; nt < 5; ++nt) acc[nt] = (v8f)(0.0f);

    for (int k = 0; k < D_; k += 4) {
        const int d = k + 2 * half;
        v2f a;
        a.x = arp[d] * rscale;
        a.y = arp[d + 1] * rscale;
        #pragma unroll
        for (int nt = 0; nt < 5; ++nt) {
            const int col = nt * 16 + l16;
            const float* wsrc = (col < G4_) ? (w_ih_f + (long)col * D_)
                                            : (w_ih_b + (long)(col - G4_) * D_);
            v2f bv;
            bv.x = wsrc[d];
            bv.y = wsrc[d + 1];
            acc[nt] = __builtin_amdgcn_wmma_f32_16x16x4_f32(
                false, a, false, bv, (short)0, acc[nt], false, false);
        }
    }

    // C/D layout: VGPR r, lanes 0-15 -> M=r ; lanes 16-31 -> M=r+8 ; N = lane&15
    #pragma unroll
    for (int nt = 0; nt < 5; ++nt) {
        const int n = nt * 16 + l16;
        const float bias = (n < G4_) ? b_f[n] : b_b[n - G4_];
        #pragma unroll
        for (int r = 0; r < 8; ++r) {
            const int m = m0 + r + half * 8;
            const int t = m & (W_ - 1);
            const int b = m >> 8;
            const float v = acc[nt][r] + bias;
            if (n < G4_) pre_f[((long)t * B_ + b) * G4_ + n] = v;
            else         pre_b[((long)t * B_ + b) * G4_ + (n - G4_)] = v;
        }
    }
}

// ---------------- K3: LSTM scan (one block per direction) ----------------
__global__ __launch_bounds__(320) void k_lstm(const float* __restrict__ pre_f,
                                              const float* __restrict__ pre_b,
                                              const float* __restrict__ w_hh_f,
                                              const float* __restrict__ w_hh_b,
                                              float* __restrict__ u) {
    const int dir = blockIdx.x;
    const float* pre = dir ? pre_b : pre_f;
    const float* whh = dir ? w_hh_b : w_hh_f;
    const int tid = threadIdx.x;

    __shared__ float hs[B_ * H_];        // 320
    __shared__ float cs[B_ * H_];
    __shared__ float gs[B_ * G4_];       // 1280
    __shared__ float wsh[G4_ * H_];      // 400

    for (int i = tid; i < G4_ * H_; i += 320) wsh[i] = whh[i];
    hs[tid] = 0.0f;
    cs[tid] = 0.0f;
    __syncthreads();

    for (int step = 0; step < W_; ++step) {
        const int t = dir ? (W_ - 1 - step) : step;
        const float* p = pre + (long)t * B_ * G4_;
        // gates: g[b,j] = pre[t,b,j] + sum_k h[b,k]*whh[j,k]
        for (int e = tid; e < B_ * G4_; e += 320) {
            const int b = e / G4_, j = e % G4_;
            float s = p[e];
            #pragma unroll
            for (int k = 0; k < H_; ++k) s += hs[b * H_ + k] * wsh[j * H_ + k];
            gs[e] = s;
        }
        __syncthreads();
        {
            const int b = tid / H_, hh = tid % H_;
            const float ig = sigmoidf_(gs[b * G4_ + hh]);
            const float fg = sigmoidf_(gs[b * G4_ + H_ + hh]);
            const float gg = tanhf(gs[b * G4_ + 2 * H_ + hh]);
            const float og = sigmoidf_(gs[b * G4_ + 3 * H_ + hh]);
            const float c  = fg * cs[tid] + ig * gg;
            const float h  = og * tanhf(c);
            cs[tid] = c;
            hs[tid] = h;
            u[((long)b * W_ + t) * 20 + dir * H_ + hh] = h;
        }
        __syncthreads();
    }
}

// ---------------- K4: biaffine projections + augment + g1u ----------------
__global__ __launch_bounds__(128) void k_proj(const float* __restrict__ u,
                                              const float* __restrict__ uW1, const float* __restrict__ ub1,
                                              const float* __restrict__ uW2, const float* __restrict__ ub2,
                                              const float* __restrict__ rW1, const float* __restrict__ rb1,
                                              const float* __restrict__ rW2, const float* __restrict__ rb2,
                                              const float* __restrict__ uU,
                                              float* __restrict__ h1u, float* __restrict__ h2u,
                                              float* __restrict__ h1r, float* __restrict__ h2r,
                                              float* __restrict__ g1u) {
    const long bt = blockIdx.x;
    const int tid = threadIdx.x;
    __shared__ float us[20];
    __shared__ float h1s[32];
    if (tid < 20) us[tid] = u[bt * 20 + tid];
    __syncthreads();

    if (tid < 120) {
        const int which = tid / HID_, unit = tid % HID_;
        const float* Wm = (which == 0) ? uW1 : (which == 1) ? uW2 : (which == 2) ? rW1 : rW2;
        const float* bv = (which == 0) ? ub1 : (which == 1) ? ub2 : (which == 2) ? rb1 : rb2;
        float s = bv[unit];
        #pragma unroll
        for (int k = 0; k < 20; ++k) s += us[k] * Wm[k * HID_ + unit];
        s = fmaxf(s, 0.0f);
        float* dst = (which == 0) ? h1u : (which == 1) ? h2u : (which == 2) ? h1r : h2r;
        dst[bt * 32 + unit] = s;
        if (which == 0) h1s[unit] = s;
    }
    if (tid < 4) {
        float* dst = (tid == 0) ? h1u : (tid == 1) ? h2u : (tid == 2) ? h1r : h2r;
        dst[bt * 32 + HID_]     = 1.0f;   // augmented 1
        dst[bt * 32 + HID_ + 1] = 0.0f;   // pad
    }
    if (tid == 0) { h1s[HID_] = 1.0f; h1s[HID_ + 1] = 0.0f; }
    __syncthreads();

    // g1u[k] = sum_h h1p_u[h] * uU[h,0,k]
    if (tid < HP_) {
        float s = 0.0f;
        #pragma unroll
        for (int h = 0; h < HP_; ++h) s += h1s[h] * uU[h * HP_ + tid];
        g1u[bt * 32 + tid] = s;
    } else if (tid == HP_) {
        g1u[bt * 32 + HP_] = 0.0f;
    }
}

// ---------------- K5: unlabeled scores + fused log-softmax NLL ----------------
__global__ __launch_bounds__(256) void k_unl_loss(const float* __restrict__ g1u,
                                                  const float* __restrict__ h2u,
                                                  const int* __restrict__ heads,
                                                  const unsigned char* __restrict__ masks,
                                                  float* __restrict__ out) {
    const int b = blockIdx.x >> 8;
    const int n = blockIdx.x & (W_ - 1);
    const int tid = threadIdx.x;
    __shared__ float g1s[32];
    __shared__ float sc[W_];
    __shared__ float red[W_];

    if (tid < 32) g1s[tid] = g1u[((long)b * W_ + n) * 32 + tid];
    __syncthreads();

    const float* h2 = h2u + ((long)b * W_ + tid) * 32;
    float s = 0.0f;
    #pragma unroll
    for (int k = 0; k < HP_; ++k) s += g1s[k] * h2[k];
    if (tid == n) s = -1.0e9f;     // diagonal mask
    sc[tid]  = s;
    red[tid] = s;
    __syncthreads();
    for (int off = 128; off > 0; off >>= 1) {
        if (tid < off) red[tid] = fmaxf(red[tid], red[tid + off]);
        __syncthreads();
    }
    const float mx = red[0];
    __syncthreads();
    red[tid] = expf(s - mx);
    __syncthreads();
    for (int off = 128; off > 0; off >>= 1) {
        if (tid < off) red[tid] += red[tid + off];
        __syncthreads();
    }
    if (tid == 0 && n >= 1) {
        if (!masks[b * W_ + n]) {            // heads_ == -1 excluded
            const int hm = heads[b * W_ + n];
            const float lse = mx + logf(red[0]);
            atomicAdd(out, -(sc[hm] - lse));
        }
    }
}

// ---------------- K6: deprel scores at head positions only ----------------
__global__ __launch_bounds__(64) void k_dep_scores(const float* __restrict__ h1r,
                                                   const float* __restrict__ h2r,
                                                   const float* __restrict__ rU,
                                                   const int* __restrict__ heads,
                                                   float* __restrict__ picked) {
    const int b = blockIdx.x >> 8;
    const int n = blockIdx.x & (W_ - 1);
    const int tid = threadIdx.x;
    __shared__ float h1s[32];
    __shared__ float h2s[32];
    const int hd = heads[b * W_ + n];
    if (tid < 32) {
        h1s[tid] = h1r[((long)b * W_ + n) * 32 + tid];
        h2s[tid] = h2r[((long)b * W_ + hd) * 32 + tid];
    }
    __syncthreads();
    if (tid < C_) {
        float acc = 0.0f;
        for (int h = 0; h < HP_; ++h) {
            const float* Ur = rU + ((long)h * C_ + tid) * HP_;
            float th = 0.0f;
            #pragma unroll
            for (int k = 0; k < HP_; ++k) th += Ur[k] * h2s[k];
            acc += h1s[h] * th;
        }
        picked[((long)b * W_ + n) * C_ + tid] = acc;
    }
}

// ---------------- K7: scrambled-reshape log-softmax NLL (deprel loss) ----------------
// head_scores[b,i,j] = picked[b, (i*256+j)/50, (i*256+j)%50]; softmax over i.
__global__ __launch_bounds__(256) void k_dep_loss(const float* __restrict__ picked,
                                                  const int* __restrict__ dep_rels,
                                                  float* __restrict__ out) {
    const int b = blockIdx.x;
    const int j = threadIdx.x;
    float v[C_];
    float mx = -3.4e38f;
    #pragma unroll
    for (int i = 0; i < C_; ++i) {
        const int k = i * W_ + j;
        const int w = k / C_, c = k % C_;
        v[i] = picked[((long)b * W_ + w) * C_ + c];
        mx = fmaxf(mx, v[i]);
    }
    float se = 0.0f;
    #pragma unroll
    for (int i = 0; i < C_; ++i) se += expf(v[i] - mx);
    const float lse = mx + logf(se);
    const int r = dep_rels[b * W_ + j];
    const float loss = (r != 0) ? -(v[r] - lse) : 0.0f;
    atomicAdd(out, loss);
}

// ---------------- host side ----------------
extern "C" void kernel_launch(void* const* d_in, const int* in_sizes, int n_in,
                              void* d_out, int out_size, void* d_ws, size_t ws_size,
                              hipStream_t stream) {
    (void)in_sizes; (void)n_in; (void)ws_size;
    const float* hiddens = (const float*)d_in[0];
    const float* w_ih_f  = (const float*)d_in[1];
    const float* w_hh_f  = (const float*)d_in[2];
    const float* b_f     = (const float*)d_in[3];
    const float* w_ih_b  = (const float*)d_in[4];
    const float* w_hh_b  = (const float*)d_in[5];
    const float* b_b     = (const float*)d_in[6];
    const float* uW1     = (const float*)d_in[7];
    const float* ub1     = (const float*)d_in[8];
    const float* uW2     = (const float*)d_in[9];
    const float* ub2     = (const float*)d_in[10];
    const float* uU      = (const float*)d_in[11];
    const float* rW1     = (const float*)d_in[12];
    const float* rb1     = (const float*)d_in[13];
    const float* rW2     = (const float*)d_in[14];
    const float* rb2     = (const float*)d_in[15];
    const float* rU      = (const float*)d_in[16];
    const int*   bert2toks = (const int*)d_in[17];
    const int*   heads     = (const int*)d_in[18];
    const int*   dep_rels  = (const int*)d_in[19];
    const unsigned char* masks = (const unsigned char*)d_in[20];
    float* out = (float*)d_out;
    float* ws  = (float*)d_ws;

    float* word  = ws + O_WORD;
    float* cnt   = ws + O_CNT;
    float* inv   = ws + O_INV;
    float* pre_f = ws + O_PREF;
    float* pre_b = ws + O_PREB;
    float* u     = ws + O_U;
    float* h1u   = ws + O_H1U;
    float* h2u   = ws + O_H2U;
    float* h1r   = ws + O_H1R;
    float* h2r   = ws + O_H2R;
    float* g1u   = ws + O_G1U;
    float* pick  = ws + O_PICK;

    // zero the accumulated regions + output
    {
        long nz = WORD_N + CNT_N;
        k_zero<<<(unsigned)((nz + 255) / 256), 256, 0, stream>>>(word, nz);
        k_zero<<<(unsigned)((out_size + 255) / 256), 256, 0, stream>>>(out, out_size);
    }
    k_segsum<<<B_ * SUB_, 256, 0, stream>>>(hiddens, bert2toks, word, cnt);
    k_inv<<<(B_ * W_ + 255) / 256, 256, 0, stream>>>(cnt, inv);
    k_pre_wmma<<<(B_ * W_) / 16, 32, 0, stream>>>(word, inv, w_ih_f, w_ih_b, b_f, b_b, pre_f, pre_b);
    k_lstm<<<2, 320, 0, stream>>>(pre_f, pre_b, w_hh_f, w_hh_b, u);
    k_proj<<<B_ * W_, 128, 0, stream>>>(u, uW1, ub1, uW2, ub2, rW1, rb1, rW2, rb2, uU,
                                        h1u, h2u, h1r, h2r, g1u);
    k_unl_loss<<<B_ * W_, 256, 0, stream>>>(g1u, h2u, heads, masks, out);
    k_dep_scores<<<B_ * W_, 64, 0, stream>>>(h1r, h2r, rU, heads, pick);
    k_dep_loss<<<B_, 256, 0, stream>>>(pick, dep_rels, out);
}